// HierarchicalMassivePool_86431921865018
// MI455X (gfx1250) — compile-verified
//
#include <hip/hip_runtime.h>

// ---------------------------------------------------------------------------
// HierarchicalMassivePool for MI455X (gfx1250, wave32, WMMA + async-to-LDS)
//   scores = (mean_T(hidden) @ W + b) @ keys^T / sqrt(D)  -> top-64 -> softmax
//   -> weighted gather of pool_params
// Hot loop: double-buffered GLOBAL_LOAD_ASYNC_TO_LDS_B128 key staging feeding
// V_WMMA_F32_16X16X4_F32, keeping the 1 GB key sweep at the HBM roofline.
// ---------------------------------------------------------------------------

typedef __attribute__((ext_vector_type(2))) float v2f;
typedef __attribute__((ext_vector_type(8))) float v8f;
typedef __attribute__((ext_vector_type(4))) int   i4;

#define D_DIM   512
#define QN      32
#define T_DIM   1024
#define NKEYS   500000
#define KPB     128                 // keys per block in the WMMA kernel
#define KS      32                  // K-slab staged in LDS (per buffer)
#define NSLAB   (D_DIM / KS)        // 16 slabs
#define SLS     36                  // padded LDS row stride (floats): no bank conflicts
#define SROWS   (KPB + QN)          // 160 rows per slab buffer (keys + queries)
#define CSTR    132                 // padded C-stage stride (floats)
#define CHUNK   8192                // keys per top-k partial block
#define NCHUNK  62                  // 62*8192 = 507904
#define NPAD    (NCHUNK * CHUNK)    // padded key count (507904)
#define NBLK_A  (NPAD / KPB)        // 3968 score blocks
#define MAXK    64
#define NCAND   (NCHUNK * MAXK)     // 3968 candidates per row

// ----------------------------------------------------- CDNA5 async helpers --
typedef __attribute__((address_space(1))) i4* gp4;   // global int4*
typedef __attribute__((address_space(3))) i4* lp4;   // LDS int4*

__device__ __forceinline__ void async_copy_b128(const float* g, float* l) {
#if defined(__has_builtin) && __has_builtin(__builtin_amdgcn_global_load_async_to_lds_b128)
  __builtin_amdgcn_global_load_async_to_lds_b128((gp4)g, (lp4)l, 0, 0);
#else
  // GVS form: mem = SGPR64 + VGPR32(signed) + IOFFSET ; VDST = LDS byte address
  unsigned lds_off = (unsigned)(__SIZE_TYPE__)((__attribute__((address_space(3))) void*)l);
  unsigned long long ga = (unsigned long long)(__SIZE_TYPE__)g;
  asm volatile("global_load_async_to_lds_b128 %0, %1, %2 offset:0"
               :: "v"(lds_off), "v"((unsigned)0), "s"(ga) : "memory");
#endif
}

#if defined(__has_builtin) && __has_builtin(__builtin_amdgcn_s_wait_asynccnt)
#define WAIT_ASYNC(n) __builtin_amdgcn_s_wait_asynccnt(n)
#else
#define WAIT_ASYNC(n) asm volatile("s_wait_asynccnt %0" :: "i"(n) : "memory")
#endif

// ---------------------------------------------------------------- kernel 0 --
__global__ __launch_bounds__(256) void pool_mean_kernel(
    const float* __restrict__ hidden, float* __restrict__ pooled) {
  const int b = blockIdx.x;
  const int d = threadIdx.x;
  const float* base = hidden + (long long)b * T_DIM * D_DIM;
  float s0 = 0.f, s1 = 0.f;
  for (int t = 0; t < T_DIM; ++t) {
    s0 += base[t * D_DIM + d];
    s1 += base[t * D_DIM + d + 256];
  }
  pooled[b * D_DIM + d]       = s0 * (1.0f / (float)T_DIM);
  pooled[b * D_DIM + d + 256] = s1 * (1.0f / (float)T_DIM);
}

// ---------------------------------------------------------------- kernel 1 --
__global__ __launch_bounds__(256) void query_proj_kernel(
    const float* __restrict__ pooled, const float* __restrict__ W,
    const float* __restrict__ bias, float* __restrict__ query) {
  const int q = blockIdx.x;
  const int d = threadIdx.x;
  float a0 = 0.f, a1 = 0.f;
  for (int k = 0; k < D_DIM; ++k) {
    const float p = pooled[q * D_DIM + k];
    a0 += p * W[k * D_DIM + d];
    a1 += p * W[k * D_DIM + d + 256];
  }
  query[q * D_DIM + d]       = a0 + bias[d];
  query[q * D_DIM + d + 256] = a1 + bias[d + 256];
}

// ---------------------------------------------------------------- kernel 2 --
// Block: 256 threads = 8 waves. Wave w owns keys [key0+16w, key0+16w+16).
// A = keys tile (16x4), B = query tile (4x16), C = 16 keys x 16 queries.
// Two accumulators per wave cover all 32 queries. Key/query K-slabs are
// DMA'd into double-buffered LDS with GLOBAL_LOAD_ASYNC_TO_LDS_B128 so the
// async unit streams slab s+1 while WMMAs chew on slab s.
__global__ __launch_bounds__(256) void scores_wmma_kernel(
    const float* __restrict__ keys, const float* __restrict__ query,
    float* __restrict__ scores) {
  __shared__ float sbuf[2][SROWS * SLS];  // 2 x 22.5 KB (keys rows 0..127, query rows 128..159)
  __shared__ float cst[QN * CSTR];        // 16.5 KB C transpose stage

  const int tid  = threadIdx.x;
  const int wave = tid >> 5;
  const int lane = tid & 31;
  const int l16  = lane & 15;
  const int hi2  = (lane >> 4) << 1;      // ISA A/B layout: lanes 16-31 hold K+2,K+3
  const long long key0 = (long long)blockIdx.x * KPB;

  // per-thread share of one slab: 4 key float4's + 1 query float4 (5 async/wave)
  auto issue_slab = [&](float* buf, int ks) {
#pragma unroll
    for (int t = 0; t < 4; ++t) {
      const int i   = tid + t * 256;      // 0..1023 over 128 rows x 8 float4
      const int row = i >> 3;
      const int c4  = i & 7;
      long long key = key0 + row;
      if (key > (long long)NKEYS - 1) key = NKEYS - 1;   // tail: clamped, fixed at writeout
      async_copy_b128(&keys[key * D_DIM + ks + c4 * 4], &buf[row * SLS + c4 * 4]);
    }
    const int qrow = tid >> 3, qc4 = tid & 7;            // 32 rows x 8 float4
    async_copy_b128(&query[qrow * D_DIM + ks + qc4 * 4],
                    &buf[(KPB + qrow) * SLS + qc4 * 4]);
  };

  v8f c_lo = {};
  v8f c_hi = {};

  issue_slab(&sbuf[0][0], 0);             // prologue: slab 0 in flight
  for (int s = 0; s < NSLAB; ++s) {
    if (s + 1 < NSLAB) {
      issue_slab(&sbuf[(s + 1) & 1][0], (s + 1) * KS);   // prefetch next slab
      WAIT_ASYNC(5);                      // in-order completion => slab s landed (this wave)
    } else {
      WAIT_ASYNC(0);
    }
    __syncthreads();                      // all waves' DMA for slab s visible

    const float* base    = &sbuf[s & 1][0];
    const float* arow    = &base[(wave * 16 + l16) * SLS];
    const float* brow_lo = &base[(KPB + l16) * SLS];
    const float* brow_hi = &base[(KPB + 16 + l16) * SLS];
#pragma unroll
    for (int kk = 0; kk < KS; kk += 4) {
      const v2f a   = *(const v2f*)&arow[kk + hi2];
      const v2f blo = *(const v2f*)&brow_lo[kk + hi2];
      const v2f bhi = *(const v2f*)&brow_hi[kk + hi2];
      c_lo = __builtin_amdgcn_wmma_f32_16x16x4_f32(false, a, false, blo,
                                                   (short)0, c_lo, false, false);
      c_hi = __builtin_amdgcn_wmma_f32_16x16x4_f32(false, a, false, bhi,
                                                   (short)0, c_hi, false, false);
    }
    __syncthreads();                      // done reading sbuf[s&1] before it is recycled
  }

  // scale by 1/sqrt(D) and transpose C through LDS (scores[q][key_local])
  const float scale = 0.04419417382415922f;   // 1/sqrt(512)
  const int mbase = (lane >= 16) ? 8 : 0;     // C layout: lanes 16-31 hold M=r+8
#pragma unroll
  for (int r = 0; r < 8; ++r) {
    const int key_local = wave * 16 + r + mbase;
    cst[l16 * CSTR + key_local]        = c_lo[r] * scale;
    cst[(l16 + 16) * CSTR + key_local] = c_hi[r] * scale;
  }
  __syncthreads();
  // coalesced writeout of the 32 x 128 tile; pad keys -> -1e30
  for (int i = tid; i < QN * (KPB / 4); i += 256) {
    const int q = i >> 5, c4 = i & 31;
    const float* src = &cst[q * CSTR + c4 * 4];
    const long long kbase = key0 + c4 * 4;
    float4 v;
    v.x = (kbase + 0 < NKEYS) ? src[0] : -1e30f;
    v.y = (kbase + 1 < NKEYS) ? src[1] : -1e30f;
    v.z = (kbase + 2 < NKEYS) ? src[2] : -1e30f;
    v.w = (kbase + 3 < NKEYS) ? src[3] : -1e30f;
    *(float4*)&scores[(long long)q * NPAD + key0 + c4 * 4] = v;
  }
}

// ---------------------------------------------------------------- kernel 3 --
// Per (chunk, row): top-64 of 8192 scores by iterative block argmax in LDS.
__global__ __launch_bounds__(256) void topk_partial_kernel(
    const float* __restrict__ scores, float* __restrict__ cand_val,
    int* __restrict__ cand_idx) {
  __shared__ float buf[CHUNK];     // 32 KB
  __shared__ float rval[256];
  __shared__ int   ridx[256];
  const int chunk = blockIdx.x;
  const int row   = blockIdx.y;
  const int tid   = threadIdx.x;
  const float* src = scores + (long long)row * NPAD + (long long)chunk * CHUNK;
  for (int i = tid; i < CHUNK / 4; i += 256)
    *(float4*)&buf[i * 4] = *(const float4*)&src[i * 4];
  __syncthreads();

  const int out_base = (row * NCHUNK + chunk) * MAXK;
  for (int it = 0; it < MAXK; ++it) {
    float lv = -3e38f; int li = 0;
    for (int j = 0; j < CHUNK / 256; ++j) {
      const int idx = j * 256 + tid;           // interleaved -> conflict-free
      const float v = buf[idx];
      if (v > lv) { lv = v; li = idx; }
    }
    rval[tid] = lv; ridx[tid] = li;
    __syncthreads();
    for (int s = 128; s > 0; s >>= 1) {
      if (tid < s && rval[tid + s] > rval[tid]) {
        rval[tid] = rval[tid + s]; ridx[tid] = ridx[tid + s];
      }
      __syncthreads();
    }
    if (tid == 0) {
      cand_val[out_base + it] = rval[0];
      cand_idx[out_base + it] = chunk * CHUNK + ridx[0];
      buf[ridx[0]] = -3e38f;
    }
    __syncthreads();
  }
}

// ---------------------------------------------------------------- kernel 4 --
// Per row: merge 3968 candidates -> sorted top-64, masked softmax, gather.
__global__ __launch_bounds__(256) void topk_final_kernel(
    const float* __restrict__ cand_val, const int* __restrict__ cand_idx,
    const float* __restrict__ pool_params, const int* __restrict__ k_dyn_p,
    float* __restrict__ out) {
  __shared__ float cv[NCAND];
  __shared__ int   ci[NCAND];
  __shared__ float rval[256];
  __shared__ int   ridx[256];
  __shared__ float sel_v[MAXK];
  __shared__ int   sel_i[MAXK];
  __shared__ float sel_w[MAXK];
  const int row = blockIdx.x;
  const int tid = threadIdx.x;
  for (int i = tid; i < NCAND; i += 256) {
    cv[i] = cand_val[row * NCAND + i];
    ci[i] = cand_idx[row * NCAND + i];
  }
  __syncthreads();

  for (int it = 0; it < MAXK; ++it) {
    float lv = -3e38f; int li = 0;
    for (int i = tid; i < NCAND; i += 256)
      if (cv[i] > lv) { lv = cv[i]; li = i; }
    rval[tid] = lv; ridx[tid] = li;
    __syncthreads();
    for (int s = 128; s > 0; s >>= 1) {
      if (tid < s && rval[tid + s] > rval[tid]) {
        rval[tid] = rval[tid + s]; ridx[tid] = ridx[tid + s];
      }
      __syncthreads();
    }
    if (tid == 0) {
      sel_v[it] = rval[0];
      sel_i[it] = ci[ridx[0]];
      cv[ridx[0]] = -3e38f;
    }
    __syncthreads();
  }

  if (tid == 0) {                     // masked softmax over sorted top-64
    int kd = *k_dyn_p;
    if (kd < 1) kd = 1;
    if (kd > MAXK) kd = MAXK;
    const float m = sel_v[0];
    float sum = 0.f;
    for (int j = 0; j < MAXK; ++j) {
      const float w = (j < kd) ? __expf(sel_v[j] - m) : 0.f;
      sel_w[j] = w; sum += w;
    }
    const float inv = 1.0f / sum;
    for (int j = 0; j < MAXK; ++j) sel_w[j] *= inv;
  }
  __syncthreads();

  for (int d = tid; d < D_DIM; d += 256) {   // weighted gather
    float acc = 0.f;
    for (int j = 0; j < MAXK; ++j) {
      const float w = sel_w[j];
      if (w != 0.f) acc += w * pool_params[(long long)sel_i[j] * D_DIM + d];
    }
    out[row * D_DIM + d] = acc;
  }
}

// --------------------------------------------------------------------------
extern "C" void kernel_launch(void* const* d_in, const int* in_sizes, int n_in,
                              void* d_out, int out_size, void* d_ws, size_t ws_size,
                              hipStream_t stream) {
  const float* hidden      = (const float*)d_in[0];
  const float* pool_params = (const float*)d_in[1];
  const float* pool_keys   = (const float*)d_in[2];
  const float* W           = (const float*)d_in[3];
  const float* bias        = (const float*)d_in[4];
  const int*   k_dyn       = (const int*)d_in[5];
  float* out = (float*)d_out;

  // workspace layout
  char* ws = (char*)d_ws;
  float* pooled   = (float*)(ws);                                   // 64 KB
  float* query    = (float*)(ws + 65536);                           // 64 KB
  float* scores   = (float*)(ws + 131072);                          // 32*507904*4 = 65 MB
  size_t sc_bytes = (size_t)QN * NPAD * sizeof(float);
  float* cand_val = (float*)(ws + 131072 + sc_bytes);               // 0.5 MB
  int*   cand_idx = (int*)((char*)cand_val + (size_t)QN * NCAND * sizeof(float));

  pool_mean_kernel<<<QN, 256, 0, stream>>>(hidden, pooled);
  query_proj_kernel<<<QN, 256, 0, stream>>>(pooled, W, bias, query);
  scores_wmma_kernel<<<NBLK_A, 256, 0, stream>>>(pool_keys, query, scores);
  topk_partial_kernel<<<dim3(NCHUNK, QN), 256, 0, stream>>>(scores, cand_val, cand_idx);
  topk_final_kernel<<<QN, 256, 0, stream>>>(cand_val, cand_idx, pool_params, k_dyn, out);
}